// GATClassifier_25881472926447
// MI455X (gfx1250) — compile-verified
//
#include <hip/hip_runtime.h>
#include <hip/hip_bf16.h>

typedef float v2f __attribute__((ext_vector_type(2)));
typedef float v8f __attribute__((ext_vector_type(8)));

// ---------------------------------------------------------------- utilities
__device__ __forceinline__ void atomicMaxF(float* addr, float val) {
    // ordered-float trick: works for mixed-sign values when init = -inf
    if (val >= 0.0f) atomicMax((int*)addr, __float_as_int(val));
    else             atomicMin((unsigned int*)addr, __float_as_uint(val));
}

// Native global_atomic_add_f32 (no CAS loop). Order-insensitive sums only.
__device__ __forceinline__ void atomicAddF(float* addr, float val) {
    unsafeAtomicAdd(addr, val);
}

__global__ void fill_f32(float* __restrict__ p, float v, size_t n) {
    size_t i = (size_t)blockIdx.x * blockDim.x + threadIdx.x;
    if (i < n) p[i] = v;
}

// ------------------------------------------------------- dual GEMM via WMMA
// outL = X @ Wl + bl ; outR = X @ Wr + br
// X: M x K (row major, f32), W: K x (NT*16), one wave per 16-row stripe,
// all NT column tiles + both weight matrices computed in-wave (A reuse).
// Uses V_WMMA_F32_16X16X4_F32 (f32 A/B/C — matches f32 reference exactly).
template<int NT>
__global__ void gemm_dual_wmma(const float* __restrict__ X,
                               const float* __restrict__ Wl, const float* __restrict__ bl,
                               const float* __restrict__ Wr, const float* __restrict__ br,
                               float* __restrict__ outL, float* __restrict__ outR,
                               int M, int K) {
    const int NF   = NT * 16;
    const int wave = blockIdx.x * (blockDim.x >> 5) + (threadIdx.x >> 5);
    const int lane = threadIdx.x & 31;
    if (wave * 16 >= M) return;              // wave-uniform: EXEC stays all-ones

    const int row   = wave * 16 + (lane & 15);   // A: lane 0-15 -> M, halves -> K pairs
    const int khalf = (lane >> 4) * 2;           // K offset 0 or 2 within the x4 step
    const int ncol  = lane & 15;                 // B/C/D column within tile

    v8f zero = {0.f,0.f,0.f,0.f,0.f,0.f,0.f,0.f};
    v8f accL[NT], accR[NT];
#pragma unroll
    for (int t = 0; t < NT; ++t) { accL[t] = zero; accR[t] = zero; }

    // per-lane A pointer: 8B-aligned (row*K, kk multiple of 4, khalf in {0,2})
    const float* xrow = X + (size_t)row * K + khalf;
    for (int kk = 0; kk < K; kk += 4) {
        v2f a = *reinterpret_cast<const v2f*>(xrow + kk);   // single b64 load
#pragma unroll
        for (int t = 0; t < NT; ++t) {
            const int col = t * 16 + ncol;
            v2f bL, bR;
            bL.x = Wl[(size_t)(kk + khalf + 0) * NF + col];
            bL.y = Wl[(size_t)(kk + khalf + 1) * NF + col];
            bR.x = Wr[(size_t)(kk + khalf + 0) * NF + col];
            bR.y = Wr[(size_t)(kk + khalf + 1) * NF + col];
            accL[t] = __builtin_amdgcn_wmma_f32_16x16x4_f32(false, a, false, bL,
                                                            (short)0, accL[t], false, false);
            accR[t] = __builtin_amdgcn_wmma_f32_16x16x4_f32(false, a, false, bR,
                                                            (short)0, accR[t], false, false);
        }
    }

    // D layout: VGPR r -> M = r (+8 for lanes 16-31), N = lane%16
    const int mbase = wave * 16 + ((lane >> 4) << 3);
#pragma unroll
    for (int t = 0; t < NT; ++t) {
        const int col = t * 16 + ncol;
        const float bLv = bl[col], bRv = br[col];
#pragma unroll
        for (int r = 0; r < 8; ++r) {
            const size_t o = (size_t)(mbase + r) * NF + col;
            outL[o] = accL[t][r] + bLv;
            outR[o] = accR[t][r] + bRv;
        }
    }
}

// ----------------------------------------------------- vectorized f32 gather
template<int C>
__device__ __forceinline__ void load_vecC(const float* __restrict__ p, float* v) {
#pragma unroll
    for (int q = 0; q < C / 4; ++q) {            // p is 16B-aligned by layout
        const float4 t = reinterpret_cast<const float4*>(p)[q];
        v[4 * q + 0] = t.x; v[4 * q + 1] = t.y;
        v[4 * q + 2] = t.z; v[4 * q + 3] = t.w;
    }
}

// ------------------------------------------------------------- edge kernels
// Edge list: ei[0..E-1] = src, ei[E..2E-1] = dst; edges [E, E+N) are self loops.
template<int H, int C>
__global__ void edge_logits(const int* __restrict__ ei, int E, int N,
                            const float* __restrict__ xl, const float* __restrict__ xr,
                            const float* __restrict__ att,
                            float* __restrict__ logit, float* __restrict__ mx) {
    long long tid = (long long)blockIdx.x * blockDim.x + threadIdx.x;
    long long tot = (long long)(E + N) * H;
    if (tid >= tot) return;
    const int h = (int)(tid % H);
    const int e = (int)(tid / H);
    int src, dst;
    if (e < E) { src = ei[e]; dst = ei[E + e]; } else { src = dst = e - E; }
    float lv[C], rv[C];
    load_vecC<C>(xl + (size_t)src * (H * C) + h * C, lv);
    load_vecC<C>(xr + (size_t)dst * (H * C) + h * C, rv);
    float acc = 0.f;
#pragma unroll
    for (int c = 0; c < C; ++c) {
        float m = lv[c] + rv[c];
        m = (m > 0.f) ? m : 0.2f * m;           // leaky_relu(., 0.2)
        acc += m * att[h * C + c];
    }
    logit[(size_t)e * H + h] = acc;
    atomicMaxF(&mx[(size_t)dst * H + h], acc);
}

template<int H>
__global__ void edge_expsum(const int* __restrict__ ei, int E, int N,
                            const float* __restrict__ mx,
                            float* __restrict__ logit, float* __restrict__ den) {
    long long tid = (long long)blockIdx.x * blockDim.x + threadIdx.x;
    long long tot = (long long)(E + N) * H;
    if (tid >= tot) return;
    const int h = (int)(tid % H);
    const int e = (int)(tid / H);
    const int dst = (e < E) ? ei[E + e] : (e - E);
    const size_t li = (size_t)e * H + h;
    const float ex = expf(logit[li] - mx[(size_t)dst * H + h]);
    logit[li] = ex;
    atomicAddF(&den[(size_t)dst * H + h], ex);
}

template<int H, int C>
__global__ void edge_aggregate(const int* __restrict__ ei, int E, int N,
                               const float* __restrict__ xl,
                               const float* __restrict__ logit,
                               const float* __restrict__ den,
                               float* __restrict__ out) {
    long long tid = (long long)blockIdx.x * blockDim.x + threadIdx.x;
    long long tot = (long long)(E + N) * H;
    if (tid >= tot) return;
    const int h = (int)(tid % H);
    const int e = (int)(tid / H);
    int src, dst;
    if (e < E) { src = ei[e]; dst = ei[E + e]; } else { src = dst = e - E; }
    const float alpha = logit[(size_t)e * H + h] /
                        (den[(size_t)dst * H + h] + 1e-16f);
    float lv[C];
    load_vecC<C>(xl + (size_t)src * (H * C) + h * C, lv);
    float* po = out + (size_t)dst * (H * C) + h * C;
#pragma unroll
    for (int c = 0; c < C; ++c) atomicAddF(&po[c], alpha * lv[c]);
}

// ------------------------------------------------------------ pointwise ops
template<bool ELU>
__global__ void bias_act(float* __restrict__ h, const float* __restrict__ bias,
                         int F, size_t n) {
    size_t i = (size_t)blockIdx.x * blockDim.x + threadIdx.x;
    if (i >= n) return;
    const int c = (int)(i % F);
    float v = h[i] + bias[c];
    if (ELU) v = (v > 0.f) ? v : (expf(v) - 1.f);
    h[i] = v;
}

// -------------------------------------------------------------- pooling/head
__global__ void pool_accum(const float* __restrict__ h2, const int* __restrict__ batch,
                           int N, float* __restrict__ psum, float* __restrict__ cnt) {
    int i = blockIdx.x * blockDim.x + threadIdx.x;
    if (i >= N) return;
    const int b = batch[i];
    float v[16];
    load_vecC<16>(h2 + (size_t)i * 16, v);
#pragma unroll
    for (int c = 0; c < 16; ++c) atomicAddF(&psum[b * 16 + c], v[c]);
    atomicAddF(&cnt[b], 1.0f);
}

__global__ void finalize(const float* __restrict__ psum, const float* __restrict__ cnt,
                         const float* __restrict__ Wc, const float* __restrict__ bc,
                         float* __restrict__ out /* [64] scores + [64*16] pooled */) {
    const int b = threadIdx.x;
    if (b >= 64) return;
    float c = cnt[b];
    c = (c < 1.0f) ? 1.0f : c;                  // clip(counts, 1.0)
    float acc = bc[0];
#pragma unroll
    for (int j = 0; j < 16; ++j) {
        const float p = psum[b * 16 + j] / c;
        out[64 + b * 16 + j] = p;               // pooled
        acc += p * Wc[j];
    }
    out[b] = acc;                               // scores
}

// ------------------------------------------------------------------ launcher
extern "C" void kernel_launch(void* const* d_in, const int* in_sizes, int n_in,
                              void* d_out, int out_size, void* d_ws, size_t ws_size,
                              hipStream_t stream) {
    const float* x     = (const float*)d_in[0];
    const int*   ei    = (const int*)  d_in[1];   // (2, E) flattened
    const int*   batch = (const int*)  d_in[2];
    const float* Wl1   = (const float*)d_in[3];
    const float* bl1   = (const float*)d_in[4];
    const float* Wr1   = (const float*)d_in[5];
    const float* br1   = (const float*)d_in[6];
    const float* att1  = (const float*)d_in[7];
    const float* bias1 = (const float*)d_in[8];
    const float* Wl2   = (const float*)d_in[9];
    const float* bl2   = (const float*)d_in[10];
    const float* Wr2   = (const float*)d_in[11];
    const float* br2   = (const float*)d_in[12];
    const float* att2  = (const float*)d_in[13];
    const float* bias2 = (const float*)d_in[14];
    const float* Wc    = (const float*)d_in[15];
    const float* bc    = (const float*)d_in[16];

    const int N = in_sizes[2];
    const int E = in_sizes[1] / 2;
    float* out = (float*)d_out;

    // ---- workspace layout (f32 elements)
    float* ws = (float*)d_ws;
    size_t o = 0;
    float* xl1    = ws + o; o += (size_t)N * 64;
    float* xr1    = ws + o; o += (size_t)N * 64;
    float* mx1    = ws + o; o += (size_t)N * 8;
    float* den1   = ws + o; o += (size_t)N * 8;
    float* h1     = ws + o; o += (size_t)N * 64;   // agg1, then ELU in-place
    float* xl2    = ws + o; o += (size_t)N * 16;
    float* xr2    = ws + o; o += (size_t)N * 16;
    float* mx2    = ws + o; o += (size_t)N;
    float* den2   = ws + o; o += (size_t)N;
    float* h2     = ws + o; o += (size_t)N * 16;   // agg2, then +bias in-place
    float* logit1 = ws + o; o += (size_t)(E + N) * 8;
    float* logit2 = ws + o; o += (size_t)(E + N);
    float* psum   = ws + o; o += 64 * 16;
    float* cnt    = ws + o; o += 64;
    (void)ws_size; (void)n_in; (void)out_size;

    const int TB = 256;
    auto blks = [](long long n, int t) { return (unsigned)((n + t - 1) / t); };
    const float NEG_INF = -__builtin_huge_valf();
    const long long tot1 = (long long)(E + N) * 8;
    const long long tot2 = (long long)(E + N);

    // ======== layer 1: GATv2(128 -> 8 heads x 8 ch, concat) ========
    {
        const int waves = (N + 15) / 16;
        gemm_dual_wmma<4><<<blks(waves, 8), TB, 0, stream>>>(x, Wl1, bl1, Wr1, br1,
                                                             xl1, xr1, N, 128);
    }
    fill_f32<<<blks((long long)N * 8, TB), TB, 0, stream>>>(mx1, NEG_INF, (size_t)N * 8);
    fill_f32<<<blks((long long)N * 8, TB), TB, 0, stream>>>(den1, 0.f, (size_t)N * 8);
    fill_f32<<<blks((long long)N * 64, TB), TB, 0, stream>>>(h1, 0.f, (size_t)N * 64);

    edge_logits<8, 8><<<blks(tot1, TB), TB, 0, stream>>>(ei, E, N, xl1, xr1, att1, logit1, mx1);
    edge_expsum<8><<<blks(tot1, TB), TB, 0, stream>>>(ei, E, N, mx1, logit1, den1);
    edge_aggregate<8, 8><<<blks(tot1, TB), TB, 0, stream>>>(ei, E, N, xl1, logit1, den1, h1);
    bias_act<true><<<blks((long long)N * 64, TB), TB, 0, stream>>>(h1, bias1, 64, (size_t)N * 64);

    // ======== layer 2: GATv2(64 -> 1 head x 16 ch) ========
    {
        const int waves = (N + 15) / 16;
        gemm_dual_wmma<1><<<blks(waves, 8), TB, 0, stream>>>(h1, Wl2, bl2, Wr2, br2,
                                                             xl2, xr2, N, 64);
    }
    fill_f32<<<blks(N, TB), TB, 0, stream>>>(mx2, NEG_INF, (size_t)N);
    fill_f32<<<blks(N, TB), TB, 0, stream>>>(den2, 0.f, (size_t)N);
    fill_f32<<<blks((long long)N * 16, TB), TB, 0, stream>>>(h2, 0.f, (size_t)N * 16);

    edge_logits<1, 16><<<blks(tot2, TB), TB, 0, stream>>>(ei, E, N, xl2, xr2, att2, logit2, mx2);
    edge_expsum<1><<<blks(tot2, TB), TB, 0, stream>>>(ei, E, N, mx2, logit2, den2);
    edge_aggregate<1, 16><<<blks(tot2, TB), TB, 0, stream>>>(ei, E, N, xl2, logit2, den2, h2);
    bias_act<false><<<blks((long long)N * 16, TB), TB, 0, stream>>>(h2, bias2, 16, (size_t)N * 16);

    // ======== mean pool per graph + linear head ========
    fill_f32<<<blks(64 * 16, TB), TB, 0, stream>>>(psum, 0.f, 64 * 16);
    fill_f32<<<1, 64, 0, stream>>>(cnt, 0.f, 64);
    pool_accum<<<blks(N, TB), TB, 0, stream>>>(h2, batch, N, psum, cnt);
    finalize<<<1, 64, 0, stream>>>(psum, cnt, Wc, bc, out);
}